// ArticulationNoiseNetwork_83562883711054
// MI455X (gfx1250) — compile-verified
//
#include <hip/hip_runtime.h>
#include <hip/hip_bf16.h>
#include <math.h>

#define NBANDS 24
#define NB     8
#define NT     800
#define NL     192000

typedef __attribute__((ext_vector_type(2))) float v2f;
typedef __attribute__((ext_vector_type(8))) float v8f;

// D = A(16x4 f32) * B(4x16 f32) + C(16x16 f32), wave32 striped.
// A/B lane layout: M/N = lane&15; VGPR0 -> K = 2*(lane>>4), VGPR1 -> K = 2*(lane>>4)+1.
// C/D: N = lane&15, M = j + 8*(lane>>4) for VGPR j.
__device__ __forceinline__ v8f wmma_f32_16x16x4(v2f a, v2f b, v8f c) {
    return __builtin_amdgcn_wmma_f32_16x16x4_f32(false, a, false, b, (short)0, c,
                                                 false, false);
}

// ---------------------------------------------------------------------------
// Prep: transpose condition [B,C,T] -> [B,T,C] (C = 128).
// ---------------------------------------------------------------------------
__global__ void transpose_ct(const float* __restrict__ in, float* __restrict__ out)
{
    int x = blockIdx.x * blockDim.x + threadIdx.x;
    if (x >= NB * 128 * NT) return;
    int t = x % NT;
    int c = (x / NT) % 128;
    int b = x / (NT * 128);
    out[((size_t)b * NT + t) * 128 + c] = in[x];
}

// ---------------------------------------------------------------------------
// Prep: pack conv weights [Cout,Cin,3] -> [3][Cin/2][Cout][2] so each WMMA A
// fragment (two consecutive ci at fixed co,k) is one aligned b64 load.
// ---------------------------------------------------------------------------
__global__ void pack_w(const float* __restrict__ w, float* __restrict__ P,
                       int Cin, int Cout)
{
    int x = blockIdx.x * blockDim.x + threadIdx.x;
    if (x >= 3 * Cin * Cout) return;
    int co = x % Cout;
    int ci = (x / Cout) % Cin;
    int k  = x / (Cout * Cin);
    P[((size_t)(k * (Cin >> 1) + (ci >> 1)) * Cout + co) * 2 + (ci & 1)] =
        w[((size_t)co * Cin + ci) * 3 + k];
}

// ---------------------------------------------------------------------------
// K=3 "same" conv over T as WMMA GEMM, t-major in/out, packed weights.
// One wave owns TWO 16(co) x 16(t) C tiles (shared B fragment).
// Boundary handling is branch-free: clamped index + arithmetic mask.
// ---------------------------------------------------------------------------
__global__ void conv_k3_wmma(const float* __restrict__ inT,  // [B, T, Cin]
                             const float* __restrict__ wP,   // [3, Cin/2, Cout, 2]
                             const float* __restrict__ bias, // [Cout]
                             float* __restrict__ outT,       // [B, T, Cout]
                             int Cin, int Cout)
{
    const int lane = threadIdx.x & 31;
    const int wid  = (blockIdx.x * blockDim.x + threadIdx.x) >> 5;
    const int half = lane >> 4;
    const int lm   = lane & 15;

    const int tilesN  = NT / 16;               // 50
    const int tilesMp = Cout / 32;             // pairs of M tiles
    const int nt  = wid % tilesN;
    const int mtp = (wid / tilesN) % tilesMp;
    const int b   = wid / (tilesN * tilesMp);

    const int t0  = nt * 16;
    const int m0  = mtp * 32;
    const int tN  = t0 + lm;
    const int mA0 = m0 + lm;
    const int mA1 = m0 + 16 + lm;

    v8f acc0, acc1;
#pragma unroll
    for (int j = 0; j < 8; ++j) {
        acc0[j] = bias[m0 + j + 8 * half];
        acc1[j] = bias[m0 + 16 + j + 8 * half];
    }

    const float* inB = inT + (size_t)b * NT * Cin;

    for (int k = 0; k < 3; ++k) {
        const int   tin  = tN + k - 1;
        const float mask = (tin >= 0 && tin < NT) ? 1.0f : 0.0f;
        const int   tc   = min(max(tin, 0), NT - 1);
        const float* inRow = inB + (size_t)tc * Cin;
        const float* wK    = wP + (size_t)k * (Cin >> 1) * Cout * 2;
#pragma unroll 4
        for (int ci0 = 0; ci0 < Cin; ci0 += 4) {
            const float* wRow = wK + (size_t)((ci0 >> 1) + half) * Cout * 2;
            v2f a0 = *(const v2f*)(wRow + 2 * mA0);
            v2f a1 = *(const v2f*)(wRow + 2 * mA1);
            v2f bf = *(const v2f*)(inRow + ci0 + 2 * half);
            bf[0] *= mask;
            bf[1] *= mask;
            acc0 = wmma_f32_16x16x4(a0, bf, acc0);
            acc1 = wmma_f32_16x16x4(a1, bf, acc1);
        }
    }

    // leaky_relu(0.1); lane's 8 C rows are contiguous in t-major output
    v8f r0, r1;
#pragma unroll
    for (int j = 0; j < 8; ++j) {
        float v0 = acc0[j]; r0[j] = v0 > 0.0f ? v0 : 0.1f * v0;
        float v1 = acc1[j]; r1[j] = v1 > 0.0f ? v1 : 0.1f * v1;
    }
    float* op = outT + ((size_t)b * NT + tN) * Cout + m0 + 8 * half;
    *(v8f*)(op)      = r0;
    *(v8f*)(op + 16) = r1;
}

// ---------------------------------------------------------------------------
// 1x1 conv (27 outs) + sigmoid, t-major input, transposed param store.
// ---------------------------------------------------------------------------
__global__ void params_kernel(const float* __restrict__ h2T, // [B,T,256]
                              const float* __restrict__ w3,  // [27,256]
                              const float* __restrict__ b3,  // [27]
                              float* __restrict__ bandT,     // [B,T,24]
                              float* __restrict__ atk,       // [B,T]
                              float* __restrict__ inten)     // [B,T]
{
    int x = blockIdx.x * blockDim.x + threadIdx.x;
    if (x >= NB * 27 * NT) return;
    int t  = x % NT;
    int co = (x / NT) % 27;
    int b  = x / (NT * 27);
    if (co == 25) return;                      // "release" is unused
    const float4* h4 = (const float4*)(h2T + ((size_t)b * NT + t) * 256);
    const float4* w4 = (const float4*)(w3 + (size_t)co * 256);
    float s = b3[co];
    for (int i = 0; i < 64; ++i) {
        float4 h = h4[i], w = w4[i];
        s += h.x * w.x + h.y * w.y + h.z * w.z + h.w * w.w;
    }
    float sg = 1.0f / (1.0f + expf(-s));
    if (co < 24)       bandT[((size_t)b * NT + t) * 24 + co] = sg;
    else if (co == 24) atk[(size_t)b * NT + t] = sg * 10.0f;
    else               inten[(size_t)b * NT + t] = sg;      // co == 26
}

// ---------------------------------------------------------------------------
// 1x1 conv (4 outs) + softmax, t-major input, transposed store.
// ---------------------------------------------------------------------------
__global__ void ntw_kernel(const float* __restrict__ gT,   // [B,T,128]
                           const float* __restrict__ w,    // [4,128]
                           const float* __restrict__ bias, // [4]
                           float* __restrict__ ntwT)       // [B,T,4]
{
    int x = blockIdx.x * blockDim.x + threadIdx.x;
    if (x >= NB * NT) return;
    int t = x % NT;
    int b = x / NT;
    const float4* g4 = (const float4*)(gT + ((size_t)b * NT + t) * 128);
    float lg[4];
#pragma unroll
    for (int c = 0; c < 4; ++c) {
        const float4* w4 = (const float4*)(w + (size_t)c * 128);
        float s = bias[c];
        for (int i = 0; i < 32; ++i) {
            float4 h = g4[i], ww = w4[i];
            s += h.x * ww.x + h.y * ww.y + h.z * ww.z + h.w * ww.w;
        }
        lg[c] = s;
    }
    float m = fmaxf(fmaxf(lg[0], lg[1]), fmaxf(lg[2], lg[3]));
    float e[4], sum = 0.0f;
#pragma unroll
    for (int c = 0; c < 4; ++c) { e[c] = expf(lg[c] - m); sum += e[c]; }
    float inv = 1.0f / sum;
#pragma unroll
    for (int c = 0; c < 4; ++c)
        ntwT[((size_t)b * NT + t) * 4 + c] = e[c] * inv;
}

// ---------------------------------------------------------------------------
// Noise-type filters (4 x 63, pad 31) mixed by interpolated softmax weights.
// White-noise tile + coefficients staged in LDS; 63x reuse per element.
// ---------------------------------------------------------------------------
__global__ void filtered_kernel(const float* __restrict__ wn,   // [B,L]
                                const float* __restrict__ ntw,  // [4,63]
                                const float* __restrict__ ntwT, // [B,T,4]
                                float* __restrict__ filt)       // [B,L]
{
    __shared__ float s_wn[320];
    __shared__ float s_c[252];
    const int tid = threadIdx.x;
    const int blocksPerB = NL / 256;           // 750 (exact)
    const int b  = blockIdx.x / blocksPerB;
    const int l0 = (blockIdx.x % blocksPerB) * 256;

    const float* wb = wn + (size_t)b * NL;
    for (int i = tid; i < 318; i += 256) {
        int   li = l0 - 31 + i;
        float m  = (li >= 0 && li < NL) ? 1.0f : 0.0f;
        int   lc = min(max(li, 0), NL - 1);
        s_wn[i] = wb[lc] * m;
    }
    if (tid < 252) s_c[tid] = ntw[tid];
    __syncthreads();

    float f0 = 0.f, f1 = 0.f, f2 = 0.f, f3 = 0.f;
#pragma unroll 9
    for (int k = 0; k < 63; ++k) {
        float v = s_wn[tid + k];
        f0 += s_c[k]       * v;
        f1 += s_c[63 + k]  * v;
        f2 += s_c[126 + k] * v;
        f3 += s_c[189 + k] * v;
    }

    const int l = l0 + tid;
    float pos = (l + 0.5f) * ((float)NT / (float)NL) - 0.5f;
    pos = fminf(fmaxf(pos, 0.0f), (float)(NT - 1));
    int   i0 = (int)floorf(pos);
    int   i1 = min(i0 + 1, NT - 1);
    float wv = pos - (float)i0;
    const float* nb = ntwT + (size_t)b * NT * 4;
    float s = 0.0f;
    s += (nb[i0 * 4 + 0] * (1.f - wv) + nb[i1 * 4 + 0] * wv) * f0;
    s += (nb[i0 * 4 + 1] * (1.f - wv) + nb[i1 * 4 + 1] * wv) * f1;
    s += (nb[i0 * 4 + 2] * (1.f - wv) + nb[i1 * 4 + 2] * wv) * f2;
    s += (nb[i0 * 4 + 3] * (1.f - wv) + nb[i1 * 4 + 3] * wv) * f3;
    filt[(size_t)b * NL + l] = s;
}

// ---------------------------------------------------------------------------
// 24-band filter bank (24 x 31, pad 15) on WMMA. fb padded to 32x32 (two
// 16-row M tiles, taps padded to K=32). filtered tile staged in LDS (block
// covers 128 samples; 8 waves x 16-sample tiles). Branch-free edge masks.
// ---------------------------------------------------------------------------
__global__ void bands_wmma_kernel(const float* __restrict__ filt,  // [B,L]
                                  const float* __restrict__ fb,    // [24,31]
                                  const float* __restrict__ bandT, // [B,T,24]
                                  float* __restrict__ shaped)      // [B,L]
{
    __shared__ float s_f[160];                 // samples l0b-15 .. l0b+143
    const int tid  = threadIdx.x;
    const int lane = tid & 31;
    const int half = lane >> 4;
    const int lm   = lane & 15;
    const int wv   = tid >> 5;                 // wave 0..7 within block

    const int blocksPerB = NL / 128;           // 1500 (exact)
    const int b   = blockIdx.x / blocksPerB;
    const int l0b = (blockIdx.x % blocksPerB) * 128;

    const float* fB = filt + (size_t)b * NL;
    if (tid < 159) {
        int   li = l0b - 15 + tid;
        float m  = (li >= 0 && li < NL) ? 1.0f : 0.0f;
        int   lc = min(max(li, 0), NL - 1);
        s_f[tid] = fB[lc] * m;
    }
    __syncthreads();

    // A fragments: a0 = bands 0..15, a1 = bands 16..31 (rows >=24, tap 31 zero)
    v2f a0[8], a1[8];
#pragma unroll
    for (int s = 0; s < 8; ++s) {
        int k   = 4 * s + 2 * half;
        int mhi = 16 + lm;
        a0[s][0] = (k     < 31) ? fb[lm * 31 + k]     : 0.0f;
        a0[s][1] = (k + 1 < 31) ? fb[lm * 31 + k + 1] : 0.0f;
        a1[s][0] = (mhi < 24 && k     < 31) ? fb[mhi * 31 + k]     : 0.0f;
        a1[s][1] = (mhi < 24 && k + 1 < 31) ? fb[mhi * 31 + k + 1] : 0.0f;
    }

    v8f acc0 = {}; v8f acc1 = {};
    const int ib = 16 * wv + lm;               // LDS index of (l + k - 15), k=0
#pragma unroll
    for (int s = 0; s < 8; ++s) {
        int idx = ib + 4 * s + 2 * half;
        v2f bf;
        bf[0] = s_f[idx];
        bf[1] = s_f[idx + 1];
        acc0 = wmma_f32_16x16x4(a0[s], bf, acc0);
        acc1 = wmma_f32_16x16x4(a1[s], bf, acc1);
    }

    // shaped[l] = sum_band band_up * bands ; C lane holds N = lm, M = j + 8*half
    const int l = l0b + 16 * wv + lm;
    float pos = (l + 0.5f) * ((float)NT / (float)NL) - 0.5f;
    pos = fminf(fmaxf(pos, 0.0f), (float)(NT - 1));
    int   i0 = (int)floorf(pos);
    int   i1 = min(i0 + 1, NT - 1);
    float wgt = pos - (float)i0;
    const float* bT = bandT + (size_t)b * NT * 24;
    float partial = 0.0f;
#pragma unroll
    for (int j = 0; j < 8; ++j) {
        int m = j + 8 * half;
        float bu = bT[i0 * 24 + m] * (1.f - wgt) + bT[i1 * 24 + m] * wgt;
        partial += acc0[j] * bu;
        int m2 = 16 + m;
        if (m2 < 24) {
            float bu2 = bT[i0 * 24 + m2] * (1.f - wgt) + bT[i1 * 24 + m2] * wgt;
            partial += acc1[j] * bu2;
        }
    }
    partial += __shfl_xor(partial, 16, 32);    // combine M halves for same N
    if (half == 0) shaped[(size_t)b * NL + l] = partial;
}

// ---------------------------------------------------------------------------
// Noise gate (attack enhancement + 5-tap reflect box) and final outputs.
// ---------------------------------------------------------------------------
__device__ __forceinline__ float interp_t(const float* __restrict__ a, float pos)
{
    pos = fminf(fmaxf(pos, 0.0f), (float)(NT - 1));
    int   i0 = (int)floorf(pos);
    int   i1 = min(i0 + 1, NT - 1);
    float w  = pos - (float)i0;
    return a[i0] * (1.0f - w) + a[i1] * w;
}

__device__ __forceinline__ float gate_raw(const float* __restrict__ intenB,
                                          const float* __restrict__ atkB, int l)
{
    const float scale = (float)NT / (float)NL;
    float cur  = interp_t(intenB, (l + 0.5f) * scale - 0.5f);
    float diff = 0.0f;
    if (l > 0) diff = cur - interp_t(intenB, (l - 0.5f) * scale - 0.5f);
    float attack   = interp_t(atkB, (l + 0.5f) * scale - 0.5f);
    float mask     = diff > 0.1f ? 1.0f : 0.0f;
    float enhanced = fminf(fmaxf(cur + mask * attack * 0.3f, 0.0f), 1.0f);
    float inc      = diff > 0.0f ? 1.0f : 0.0f;
    return cur * (1.0f - inc) + enhanced * inc;
}

__global__ void gate_kernel(const float* __restrict__ shaped,
                            const float* __restrict__ inten,
                            const float* __restrict__ atk,
                            float* __restrict__ out)   // [B*L] shaped*gate | [B*L] gate
{
    int x = blockIdx.x * blockDim.x + threadIdx.x;
    if (x >= NB * NL) return;
    int l = x % NL;
    int b = x / NL;
    const float* iB = inten + (size_t)b * NT;
    const float* aB = atk   + (size_t)b * NT;
    float s = 0.0f;
#pragma unroll
    for (int j = -2; j <= 2; ++j) {
        int lj = l + j;
        if (lj < 0)   lj = -lj;                 // jnp reflect: -1->1, -2->2
        if (lj >= NL) lj = 2 * NL - 2 - lj;     // L->L-2, L+1->L-3
        s += gate_raw(iB, aB, lj);
    }
    float gate = s * 0.2f;
    out[(size_t)b * NL + l] = shaped[(size_t)b * NL + l] * gate;
    out[(size_t)NB * NL + (size_t)b * NL + l] = gate;
}

// ---------------------------------------------------------------------------
extern "C" void kernel_launch(void* const* d_in, const int* in_sizes, int n_in,
                              void* d_out, int out_size, void* d_ws, size_t ws_size,
                              hipStream_t stream)
{
    (void)in_sizes; (void)n_in; (void)out_size; (void)ws_size;

    const float* condition = (const float*)d_in[0];   // [8,128,800]
    const float* wn        = (const float*)d_in[1];   // [8,1,192000]
    const float* np_w1     = (const float*)d_in[2];   // [256,128,3]
    const float* np_b1     = (const float*)d_in[3];
    const float* np_w2     = (const float*)d_in[4];   // [256,256,3]
    const float* np_b2     = (const float*)d_in[5];
    const float* np_w3     = (const float*)d_in[6];   // [27,256,1]
    const float* np_b3     = (const float*)d_in[7];
    const float* ss_w1     = (const float*)d_in[8];   // [128,128,3]
    const float* ss_b1     = (const float*)d_in[9];
    const float* ss_w2     = (const float*)d_in[10];  // [4,128,1]
    const float* ss_b2     = (const float*)d_in[11];
    const float* fb_w      = (const float*)d_in[12];  // [24,1,31]
    const float* nt_w      = (const float*)d_in[13];  // [4,1,63]

    // Workspace (floats). filtered aliases h2T, shaped aliases h1T (producers
    // are dead by the time the aliases are written).
    float* ws    = (float*)d_ws;
    float* h1T   = ws;                         // [B,T,256] 1,638,400
    float* h2T   = h1T   + 1638400;            // [B,T,256] 1,638,400
    float* gT    = h2T   + 1638400;            // [B,T,128]   819,200
    float* condT = gT    + 819200;             // [B,T,128]   819,200
    float* bandT = condT + 819200;             // [B,T,24]    153,600
    float* atk   = bandT + 153600;             //               6,400
    float* inten = atk   + 6400;               //               6,400
    float* ntwT  = inten + 6400;               // [B,T,4]      25,600
    float* P1    = ntwT  + 25600;              // pack w1      98,304
    float* P2    = P1    + 98304;              // pack w2     196,608
    float* Pss   = P2    + 196608;             // pack ss_w1   49,152
    float* filtered = h2T;                     // [8, 192000]
    float* shaped   = h1T;                     // [8, 192000]

    // Prep: transpose condition, pack weights
    transpose_ct<<<(NB * 128 * NT + 255) / 256, 256, 0, stream>>>(condition, condT);
    pack_w<<<(3 * 128 * 256 + 255) / 256, 256, 0, stream>>>(np_w1, P1, 128, 256);
    pack_w<<<(3 * 256 * 256 + 255) / 256, 256, 0, stream>>>(np_w2, P2, 256, 256);
    pack_w<<<(3 * 128 * 128 + 255) / 256, 256, 0, stream>>>(ss_w1, Pss, 128, 128);

    // Frame-rate GEMM convs on WMMA (wave per pair of 16x16 tiles; exact grids)
    conv_k3_wmma<<<(NB * 8 * 50 * 32) / 256, 256, 0, stream>>>(condT, P1, np_b1, h1T, 128, 256);
    conv_k3_wmma<<<(NB * 8 * 50 * 32) / 256, 256, 0, stream>>>(h1T, P2, np_b2, h2T, 256, 256);

    params_kernel<<<(NB * 27 * NT + 255) / 256, 256, 0, stream>>>(h2T, np_w3, np_b3, bandT, atk, inten);

    conv_k3_wmma<<<(NB * 4 * 50 * 32) / 256, 256, 0, stream>>>(condT, Pss, ss_b1, gT, 128, 128);
    ntw_kernel<<<(NB * NT + 255) / 256, 256, 0, stream>>>(gT, ss_w2, ss_b2, ntwT);

    filtered_kernel<<<NB * (NL / 256), 256, 0, stream>>>(wn, nt_w, ntwT, filtered);

    // Filter bank on WMMA: 8 batches * 1500 blocks (8 waves x 16-sample tiles)
    bands_wmma_kernel<<<NB * (NL / 128), 256, 0, stream>>>(filtered, fb_w, bandT, shaped);

    gate_kernel<<<(NB * NL + 255) / 256, 256, 0, stream>>>(shaped, inten, atk, (float*)d_out);
}